// Reservoir_90323162235095
// MI455X (gfx1250) — compile-verified
//
#include <hip/hip_runtime.h>
#include <math.h>

typedef __attribute__((ext_vector_type(2))) float v2f;
typedef __attribute__((ext_vector_type(4))) float v4f;
typedef __attribute__((ext_vector_type(8))) float v8f;

#define T_STEPS 8192
#define N_RES   2048
#define DIN     64
#define DOUT    64
#define LEAK    0.3f
#define NOISEF  0.01f

#define NB   64                 // persistent blocks (<< WGP count -> safe residency)
#define TPB  256                // 8 waves of 32
#define RPB  (N_RES / NB)       // 32 rows per block
#define RPW  (RPB / (TPB/32))   // 4 rows per wave

// ---------------------------------------------------------------------------
// init: zero state double-buffer + barrier counters (runs every call)
// ---------------------------------------------------------------------------
__global__ void res_init(float* sbuf, unsigned* bar) {
    for (int i = threadIdx.x; i < 2 * N_RES; i += blockDim.x) sbuf[i] = 0.0f;
    if (threadIdx.x < 2) bar[threadIdx.x] = 0u;
}

// ---------------------------------------------------------------------------
// drive = u @ W_in^T + 0.01*noise   via v_wmma_f32_16x16x4_f32
// one wave per 16x16 tile; M-tiles=512, N-tiles=128
// ---------------------------------------------------------------------------
__global__ void drive_gemm(const float* __restrict__ u,
                           const float* __restrict__ noise,
                           const float* __restrict__ W_in,
                           float* __restrict__ drive) {
    const int wave = blockIdx.x * (blockDim.x >> 5) + (threadIdx.x >> 5);
    const int tm = wave >> 7;          // / 128
    const int tn = wave & 127;
    const int lane = threadIdx.x & 31;
    const int hi   = lane >> 4;        // 0: K pair {0,1}, 1: K pair {2,3}
    const int lm   = lane & 15;

    const float* ap = u    + (size_t)(tm * 16 + lm) * DIN + hi * 2;  // A[m][k..k+1]
    const float* bp = W_in + (size_t)(tn * 16 + lm) * DIN + hi * 2;  // B[k][n] = W_in[n][k]

    v8f c = {};
#pragma unroll
    for (int k0 = 0; k0 < DIN; k0 += 4) {
        v2f a = *(const v2f*)(ap + k0);
        v2f b = *(const v2f*)(bp + k0);
        c = __builtin_amdgcn_wmma_f32_16x16x4_f32(false, a, false, b,
                                                  (short)0, c, false, false);
    }
#pragma unroll
    for (int v = 0; v < 8; ++v) {
        const int r   = tm * 16 + hi * 8 + v;
        const int col = tn * 16 + lm;
        const size_t idx = (size_t)r * N_RES + col;
        drive[idx] = c[v] + NOISEF * noise[idx];
    }
}

// ---------------------------------------------------------------------------
// persistent recurrence: 8192 steps, grid-wide software barrier per step.
// Overwrites drive[t] with states[t] in place.
// ---------------------------------------------------------------------------
__global__ void reservoir_steps(const float* __restrict__ W,
                                float* __restrict__ ds,      // drive in, states out
                                float* __restrict__ sbuf,    // 2*N double buffer
                                unsigned* __restrict__ bar) {
    __shared__ __align__(16) float s_lds[N_RES];
    const int tid  = threadIdx.x;
    const int lane = tid & 31;
    const int wave = tid >> 5;
    const int r0   = blockIdx.x * RPB + wave * RPW;   // 4 consecutive rows
    unsigned* cnt = bar;
    unsigned* gen = bar + 1;

    const float* w0 = W + (size_t)(r0 + 0) * N_RES;
    const float* w1 = W + (size_t)(r0 + 1) * N_RES;
    const float* w2 = W + (size_t)(r0 + 2) * N_RES;
    const float* w3 = W + (size_t)(r0 + 3) * N_RES;

    for (int t = 0; t < T_STEPS; ++t) {
        const float* s_cur = sbuf + (t & 1) * N_RES;
        float*       s_nxt = sbuf + ((t + 1) & 1) * N_RES;

        for (int i = tid; i < N_RES; i += TPB) s_lds[i] = s_cur[i];
        __syncthreads();

        float a0 = 0.f, a1 = 0.f, a2 = 0.f, a3 = 0.f;
#pragma unroll 4
        for (int i = 0; i < N_RES / 128; ++i) {       // lane covers k = 4*lane + 128*i
            const int k = (lane << 2) + (i << 7);
            v4f sv = *(const v4f*)&s_lds[k];
            v4f x0 = *(const v4f*)(w0 + k);
            v4f x1 = *(const v4f*)(w1 + k);
            v4f x2 = *(const v4f*)(w2 + k);
            v4f x3 = *(const v4f*)(w3 + k);
            a0 += x0.x * sv.x + x0.y * sv.y + x0.z * sv.z + x0.w * sv.w;
            a1 += x1.x * sv.x + x1.y * sv.y + x1.z * sv.z + x1.w * sv.w;
            a2 += x2.x * sv.x + x2.y * sv.y + x2.z * sv.z + x2.w * sv.w;
            a3 += x3.x * sv.x + x3.y * sv.y + x3.z * sv.z + x3.w * sv.w;
        }
        // wave32 butterfly reduction
#pragma unroll
        for (int off = 16; off > 0; off >>= 1) {
            a0 += __shfl_xor(a0, off);
            a1 += __shfl_xor(a1, off);
            a2 += __shfl_xor(a2, off);
            a3 += __shfl_xor(a3, off);
        }
        if (lane == 0) {
            float* dst = ds + (size_t)t * N_RES;
            const float y[4] = {a0, a1, a2, a3};
#pragma unroll
            for (int j = 0; j < 4; ++j) {
                const int r = r0 + j;
                const float sn = (1.0f - LEAK) * s_lds[r]
                               + LEAK * tanhf(dst[r] + y[j]);
                dst[r]   = sn;   // states[t][r] overwrites drive[t][r]
                s_nxt[r] = sn;
            }
        }
        __threadfence();          // publish this block's state rows (agent scope)
        __syncthreads();
        if (tid == 0) {
            unsigned old = __hip_atomic_fetch_add(cnt, 1u, __ATOMIC_ACQ_REL,
                                                  __HIP_MEMORY_SCOPE_AGENT);
            if (old == NB - 1) {
                __hip_atomic_store(cnt, 0u, __ATOMIC_RELAXED,
                                   __HIP_MEMORY_SCOPE_AGENT);
                __hip_atomic_fetch_add(gen, 1u, __ATOMIC_ACQ_REL,
                                       __HIP_MEMORY_SCOPE_AGENT);
            } else {
                while (__hip_atomic_load(gen, __ATOMIC_ACQUIRE,
                                         __HIP_MEMORY_SCOPE_AGENT) < (unsigned)(t + 1)) {}
            }
        }
        __syncthreads();
        __threadfence();          // acquire other blocks' state rows
    }
}

// ---------------------------------------------------------------------------
// out = states @ w_out^T + b_out   via v_wmma_f32_16x16x4_f32
// M-tiles=512, N-tiles=4
// ---------------------------------------------------------------------------
__global__ void out_gemm(const float* __restrict__ states,
                         const float* __restrict__ w_out,
                         const float* __restrict__ b_out,
                         float* __restrict__ out) {
    const int wave = blockIdx.x * (blockDim.x >> 5) + (threadIdx.x >> 5);
    const int tm = wave >> 2;
    const int tn = wave & 3;
    const int lane = threadIdx.x & 31;
    const int hi   = lane >> 4;
    const int lm   = lane & 15;

    const float* ap = states + (size_t)(tm * 16 + lm) * N_RES + hi * 2;
    const float* bp = w_out  + (size_t)(tn * 16 + lm) * N_RES + hi * 2;  // B[k][n] = w_out[n][k]

    v8f c = {};
#pragma unroll 8
    for (int k0 = 0; k0 < N_RES; k0 += 4) {
        v2f a = *(const v2f*)(ap + k0);
        v2f b = *(const v2f*)(bp + k0);
        c = __builtin_amdgcn_wmma_f32_16x16x4_f32(false, a, false, b,
                                                  (short)0, c, false, false);
    }
#pragma unroll
    for (int v = 0; v < 8; ++v) {
        const int r   = tm * 16 + hi * 8 + v;
        const int col = tn * 16 + lm;
        out[(size_t)r * DOUT + col] = c[v] + b_out[col];
    }
}

// ---------------------------------------------------------------------------
extern "C" void kernel_launch(void* const* d_in, const int* in_sizes, int n_in,
                              void* d_out, int out_size, void* d_ws, size_t ws_size,
                              hipStream_t stream) {
    const float* u     = (const float*)d_in[0];   // (T, DIN)
    const float* noise = (const float*)d_in[1];   // (T, N)
    const float* W_in  = (const float*)d_in[2];   // (N, DIN)
    const float* W     = (const float*)d_in[3];   // (N, N)
    const float* w_out = (const float*)d_in[4];   // (DOUT, N)
    const float* b_out = (const float*)d_in[5];   // (DOUT,)
    float* out = (float*)d_out;                   // (T, DOUT)

    char* ws = (char*)d_ws;
    float* drive_states = (float*)ws;                                  // T*N f32 (64 MB)
    float* sbuf = (float*)(ws + (size_t)T_STEPS * N_RES * sizeof(float));
    unsigned* bar = (unsigned*)(ws + (size_t)T_STEPS * N_RES * sizeof(float)
                                   + (size_t)2 * N_RES * sizeof(float));

    res_init<<<1, 256, 0, stream>>>(sbuf, bar);

    // 512 M-tiles * 128 N-tiles = 65536 waves, 8 waves/block
    drive_gemm<<<(512 * 128) / 8, 256, 0, stream>>>(u, noise, W_in, drive_states);

    reservoir_steps<<<NB, TPB, 0, stream>>>(W, drive_states, sbuf, bar);

    // 512 M-tiles * 4 N-tiles = 2048 waves, 8 waves/block
    out_gemm<<<(512 * 4) / 8, 256, 0, stream>>>(drive_states, w_out, b_out, out);
}